// Head_13846974562454
// MI455X (gfx1250) — compile-verified
//
#include <hip/hip_runtime.h>
#include <math.h>

// ---------------------------------------------------------------------------
// NTM addressing head for MI455X (gfx1250, wave32).
// B=512, N=2048, F=128. HBM-bound: ~537MB traffic -> ~23us floor @ 23.3TB/s.
// One workgroup per batch keeps softmax/shift/sharpen entirely on-chip.
// m is streamed HBM -> LDS with GLOBAL_LOAD_ASYNC_TO_LDS_B128 (ASYNCcnt),
// double-buffered in 16x64 half-tiles so DMA overlaps the WMMA pipe.
// Dot products use V_WMMA_F32_16X16X4_F32 (exact fp32, matches reference).
// ---------------------------------------------------------------------------

typedef __attribute__((ext_vector_type(2))) float v2f;
typedef __attribute__((ext_vector_type(8))) float v8f;

#define BATCH   512
#define NMEM    2048
#define FDIM    128
#define EPS     1e-6f
#define COSEPS  1e-8f
#define NWAVES  4
#define TPB     (NWAVES * 32)                 // 128 threads
#define HSTRIDE 68                            // 64 + 4 pad: 16B-aligned rows, bank spread
#define HBYTES  (16 * HSTRIDE * 4)            // one half-tile buffer = 4352 B
#define TILES_PER_WAVE (NMEM / (NWAVES * 16)) // 32 tiles of 16 rows per wave
#define NSTEPS  (TILES_PER_WAVE * 2)          // 64 half-tile pipeline steps
#define NPT     (NMEM / TPB)                  // 16 elements per thread in N-phases

__device__ __forceinline__ float waveReduceAdd(float v) {
#pragma unroll
  for (int off = 16; off > 0; off >>= 1) v += __shfl_xor(v, off, 32);
  return v;
}
__device__ __forceinline__ float waveReduceMax(float v) {
#pragma unroll
  for (int off = 16; off > 0; off >>= 1) v = fmaxf(v, __shfl_xor(v, off, 32));
  return v;
}

__launch_bounds__(TPB)
__global__ void ntm_head_kernel(const float* __restrict__ m,
                                const float* __restrict__ kkey,
                                const float* __restrict__ bstr,
                                const float* __restrict__ gate,
                                const float* __restrict__ shf,
                                const float* __restrict__ shrp,
                                const float* __restrict__ w_prev,
                                float* __restrict__ out)
{
  __shared__ __align__(16) float stage[NWAVES][2][16 * HSTRIDE]; // double-buffered half-tiles
  __shared__ float simLDS[NMEM];                                 // logits -> probs -> w
  __shared__ float ldsK[FDIM];                                   // key + eps
  __shared__ float nrmLDS[NWAVES][32];                           // row-norm partials
  __shared__ float red[NWAVES];                                  // block reductions

  const int b    = blockIdx.x;
  const int tid  = threadIdx.x;
  const int wave = tid >> 5;
  const int lane = tid & 31;
  const int half = lane >> 4;   // 0: K-cols {0,1}; 1: K-cols {2,3} within each K4 chunk
  const int l16  = lane & 15;

  // ---- key (+eps) into LDS; key norm via wave32 shuffle reduce -------------
  for (int f = tid; f < FDIM; f += TPB) ldsK[f] = kkey[(size_t)b * FDIM + f] + EPS;
  __syncthreads();

  float kn = 0.f;
  for (int f = lane; f < FDIM; f += 32) { float v = ldsK[f]; kn += v * v; }
  kn = waveReduceAdd(kn);
  const float knorm = sqrtf(kn);
  const float bval  = bstr[b];

  const float* mbase = m + (size_t)b * NMEM * FDIM;
  const uint32_t stBase = (uint32_t)(uintptr_t)&stage[wave][0][0]; // LDS byte addr (addr[31:0])

  // Issue one 16x64 half-tile as 8 async B128 transfers (512B each, HBM -> LDS).
  auto issueHalf = [&](int s) {
    const int t = s >> 1, hk = s & 1;
    const int row0 = (wave * TILES_PER_WAVE + t) * 16;
    const float* srcf = mbase + (size_t)row0 * FDIM + hk * 64;
    const uint32_t lbase = stBase + (uint32_t)((s & 1) * HBYTES);
#pragma unroll
    for (int it = 0; it < 8; ++it) {
      const int flat = it * 32 + lane;    // 256 float4 slots in the half-tile
      const int rr   = flat >> 4;         // 16 float4 per 64-col row
      const int c4   = flat & 15;
      uint64_t ga = (uint64_t)(uintptr_t)(srcf + rr * FDIM + c4 * 4);
      uint32_t la = lbase + (uint32_t)(rr * (HSTRIDE * 4) + c4 * 16);
      asm volatile("global_load_async_to_lds_b128 %0, %1, off"
                   :: "v"(la), "v"(ga) : "memory");
    }
  };

  // ---- content addressing: async-stream m, WMMA dot products ---------------
  v8f  c = {};
  float npart = 0.f;
  issueHalf(0);

#pragma unroll 2
  for (int s = 0; s < NSTEPS; ++s) {
    // Prefetch next half-tile into the alternate buffer, then wait until only
    // those 8 newest transfers remain in flight (async loads retire in order).
    if (s + 1 < NSTEPS) {
      issueHalf(s + 1);
      asm volatile("s_wait_asynccnt 0x8" ::: "memory");
    } else {
      asm volatile("s_wait_asynccnt 0x0" ::: "memory");
    }

    if ((s & 1) == 0) {            // new tile: reset accumulators
#pragma unroll
      for (int v = 0; v < 8; ++v) c[v] = 0.f;
      npart = 0.f;
    }

    const int hk = s & 1;
    const float* st = &stage[wave][hk][0];
#pragma unroll
    for (int j = 0; j < 16; ++j) {
      const int fb = j * 4 + (half << 1);           // local col within the half
      v2f a = *(const v2f*)&st[l16 * HSTRIDE + fb];
      a.x += EPS; a.y += EPS;                       // me = m + eps at read time
      v2f bb = *(const v2f*)&ldsK[hk * 64 + fb];
      npart += a.x * a.x + a.y * a.y;               // row-norm partial (co-exec)
      c = __builtin_amdgcn_wmma_f32_16x16x4_f32(
              false, a, false, bb, (short)0, c, false, false);
    }

    if (s & 1) {                   // tile complete: emit 16 logits
      const int row0 = (wave * TILES_PER_WAVE + (s >> 1)) * 16;
      nrmLDS[wave][lane] = npart;
      asm volatile("s_wait_dscnt 0" ::: "memory");

      float cother[8];
#pragma unroll
      for (int v = 0; v < 8; ++v) cother[v] = __shfl_xor(c[v], 16, 32);
      float numv = 0.f;
#pragma unroll
      for (int v = 0; v < 8; ++v)
        numv = ((lane & 7) == v) ? ((lane < 8) ? c[v] : cother[v]) : numv;

      if (lane < 16) {
        float n2    = nrmLDS[wave][lane] + nrmLDS[wave][lane + 16];
        float denom = fmaxf(sqrtf(n2) * knorm, COSEPS);
        simLDS[row0 + lane] = bval * (numv / denom);   // beta * cosine sim
      }
    }
  }
  __syncthreads();

  // ---- softmax over N (block-wide) ----------------------------------------
  float mx = -3.4e38f;
#pragma unroll
  for (int j = 0; j < NPT; ++j) mx = fmaxf(mx, simLDS[j * TPB + tid]);
  mx = waveReduceMax(mx);
  if (lane == 0) red[wave] = mx;
  __syncthreads();
  mx = fmaxf(fmaxf(red[0], red[1]), fmaxf(red[2], red[3]));
  __syncthreads();

  float sm = 0.f;
#pragma unroll
  for (int j = 0; j < NPT; ++j) {
    const int n = j * TPB + tid;
    float e = __expf(simLDS[n] - mx);
    simLDS[n] = e;
    sm += e;
  }
  sm = waveReduceAdd(sm);
  if (lane == 0) red[wave] = sm;
  __syncthreads();
  const float tot = red[0] + red[1] + red[2] + red[3];
  __syncthreads();

  // ---- interpolation: w = softmax + (1-g)*w_prev --------------------------
  const float og = 1.f - gate[b];
#pragma unroll
  for (int j = 0; j < NPT; ++j) {
    const int n = j * TPB + tid;
    simLDS[n] = simLDS[n] / tot + og * w_prev[(size_t)b * NMEM + n];
  }
  __syncthreads();

  // ---- circular shift (3 taps) + sharpen + normalize ----------------------
  const float s0 = shf[b * 3 + 0], s1 = shf[b * 3 + 1], s2 = shf[b * 3 + 2];
  const float rv = shrp[b];
  float wr[NPT];
  float acc = 0.f;
#pragma unroll
  for (int j = 0; j < NPT; ++j) {
    const int n   = j * TPB + tid;
    const int nm  = (n + NMEM - 1) & (NMEM - 1);
    const int np1 = (n + 1) & (NMEM - 1);
    float ws = s0 * simLDS[nm] + s1 * simLDS[n] + s2 * simLDS[np1];
    float p  = powf(ws, rv);        // ws > 0 guaranteed
    wr[j] = p;
    acc += p;
  }
  acc = waveReduceAdd(acc);
  if (lane == 0) red[wave] = acc;
  __syncthreads();
  const float tot2 = red[0] + red[1] + red[2] + red[3] + EPS;

#pragma unroll
  for (int j = 0; j < NPT; ++j) {
    const int n = j * TPB + tid;
    out[(size_t)b * NMEM + n] = wr[j] / tot2;
  }
}

extern "C" void kernel_launch(void* const* d_in, const int* in_sizes, int n_in,
                              void* d_out, int out_size, void* d_ws, size_t ws_size,
                              hipStream_t stream) {
  const float* m  = (const float*)d_in[0];  // [B,N,F]
  const float* k  = (const float*)d_in[1];  // [B,F]
  const float* bs = (const float*)d_in[2];  // [B,1]
  const float* g  = (const float*)d_in[3];  // [B,1]
  const float* s  = (const float*)d_in[4];  // [B,3]
  const float* r  = (const float*)d_in[5];  // [B,1]
  const float* wp = (const float*)d_in[6];  // [B,N]
  float* out = (float*)d_out;               // [B,N]

  ntm_head_kernel<<<dim3(BATCH), dim3(TPB), 0, stream>>>(m, k, bs, g, s, r, wp, out);
}